// OrderCourierHeteroGNN_10333691314541
// MI455X (gfx1250) — compile-verified
//
#include <hip/hip_runtime.h>

typedef __attribute__((ext_vector_type(2))) float v2f;
typedef __attribute__((ext_vector_type(8))) float v8f;

#define N_ORDER 100000
#define N_RIDER 10000
#define N_EDGE  2000000
#define D_IN    64
#define HID     128
#define NEG_SLOPE 0.2f

__device__ __forceinline__ float leaky(float x) {
    return x > 0.f ? x : NEG_SLOPE * x;
}

// Monotonic-int encoding float atomic max (always compiles, maps to L2 atomics).
__device__ __forceinline__ void atomic_max_float(float* addr, float v) {
    if (v >= 0.f) atomicMax((int*)addr, __float_as_int(v));
    else          atomicMin((unsigned int*)addr, __float_as_uint(v));
}

// ---------------- init: zero accumulators, amax=-inf, rider_emb=bias ----------------
__global__ void init_kernel(float* __restrict__ a_src, float* __restrict__ a_dst,
                            float* __restrict__ amax, float* __restrict__ esum,
                            float* __restrict__ remb, const float* __restrict__ bias) {
    int i = blockIdx.x * 256 + threadIdx.x;
    if (i < N_RIDER * HID) remb[i] = bias[i & (HID - 1)];
    if (i < N_ORDER) a_src[i] = 0.f;
    if (i < N_RIDER) { a_dst[i] = 0.f; esum[i] = 0.f; amax[i] = -__builtin_inff(); }
}

// ---------------- order GEMM: h_src & order_proj via WMMA f32 16x16x4, fused a_src ---
// Wave layout per the CDNA5 ISA:
//   A 16x4 (MxK):  lanes 0-15 -> M=lane, K={0,1} in v0/v1; lanes 16-31 -> K={2,3}
//   B 4x16 (KxN):  lanes 0-15 -> N=lane, K={0,1};          lanes 16-31 -> K={2,3}
//   C/D 16x16:     vgpr j -> M = j + 8*(lane>=16), N = lane&15
__global__ void __launch_bounds__(256) order_gemm_kernel(
    const float* __restrict__ x,      // [N_ORDER, 64]
    const float* __restrict__ Wsrc,   // [128, 64]
    const float* __restrict__ Wproj,  // [128, 64]
    const float* __restrict__ atts,   // [128]
    const float* __restrict__ bproj,  // [128]
    float* __restrict__ h_src,        // [N_ORDER, 128]
    float* __restrict__ oproj,        // [N_ORDER, 128]
    float* __restrict__ a_src)        // [N_ORDER]
{
    const int m0    = blockIdx.x * 16;
    const int lane  = threadIdx.x & 31;
    const int wv    = threadIdx.x >> 5;     // 0..7 -> N tile
    const int h0    = wv * 16;
    const int lrow  = lane & 15;
    const int khalf = (lane >> 4) << 1;     // 0 or 2

    const float* xr  = x     + (size_t)(m0 + lrow) * D_IN;
    const float* w1r = Wsrc  + (size_t)(h0 + lrow) * D_IN;
    const float* w2r = Wproj + (size_t)(h0 + lrow) * D_IN;

    v8f acc1 = {};
    v8f acc2 = {};
#pragma unroll
    for (int k = 0; k < D_IN; k += 4) {
        v2f a, b1, b2;
        a.x  = xr[k + khalf];  a.y  = xr[k + khalf + 1];
        b1.x = w1r[k + khalf]; b1.y = w1r[k + khalf + 1];
        b2.x = w2r[k + khalf]; b2.y = w2r[k + khalf + 1];
        acc1 = __builtin_amdgcn_wmma_f32_16x16x4_f32(false, a, false, b1,
                                                     (short)0, acc1, false, false);
        acc2 = __builtin_amdgcn_wmma_f32_16x16x4_f32(false, a, false, b2,
                                                     (short)0, acc2, false, false);
    }

    const int   rbase = m0 + ((lane >> 4) << 3);
    const int   col   = h0 + lrow;
    const float att   = atts[col];
    const float bp    = bproj[col];

    float ap[8];
#pragma unroll
    for (int j = 0; j < 8; ++j) {
        float hv = acc1[j];
        h_src[(size_t)(rbase + j) * HID + col] = hv;
        oproj[(size_t)(rbase + j) * HID + col] = acc2[j] + bp;
        ap[j] = hv * att;
    }
    // reduce across the 16 lanes of each half (wave32 shfl_xor, stays in-half)
#pragma unroll
    for (int j = 0; j < 8; ++j) {
        float v = ap[j];
        v += __shfl_xor(v, 1, 32);
        v += __shfl_xor(v, 2, 32);
        v += __shfl_xor(v, 4, 32);
        v += __shfl_xor(v, 8, 32);
        if (lrow == 0) atomicAdd(&a_src[rbase + j], v);  // 8 waves x 16-col partials
    }
}

// ---------------- rider GEMM: only a_dst is needed downstream ----------------
__global__ void __launch_bounds__(256) rider_gemm_kernel(
    const float* __restrict__ x,      // [N_RIDER, 64]
    const float* __restrict__ Wdst,   // [128, 64]
    const float* __restrict__ attd,   // [128]
    float* __restrict__ a_dst)        // [N_RIDER]
{
    const int m0    = blockIdx.x * 16;
    const int lane  = threadIdx.x & 31;
    const int wv    = threadIdx.x >> 5;
    const int h0    = wv * 16;
    const int lrow  = lane & 15;
    const int khalf = (lane >> 4) << 1;

    const float* xr = x    + (size_t)(m0 + lrow) * D_IN;
    const float* wr = Wdst + (size_t)(h0 + lrow) * D_IN;

    v8f acc = {};
#pragma unroll
    for (int k = 0; k < D_IN; k += 4) {
        v2f a, b;
        a.x = xr[k + khalf]; a.y = xr[k + khalf + 1];
        b.x = wr[k + khalf]; b.y = wr[k + khalf + 1];
        acc = __builtin_amdgcn_wmma_f32_16x16x4_f32(false, a, false, b,
                                                    (short)0, acc, false, false);
    }

    const int   rbase = m0 + ((lane >> 4) << 3);
    const float att   = attd[h0 + lrow];
#pragma unroll
    for (int j = 0; j < 8; ++j) {
        float v = acc[j] * att;
        v += __shfl_xor(v, 1, 32);
        v += __shfl_xor(v, 2, 32);
        v += __shfl_xor(v, 4, 32);
        v += __shfl_xor(v, 8, 32);
        if (lrow == 0) atomicAdd(&a_dst[rbase + j], v);
    }
}

// ---------------- edge pass A: segment max of leaky(alpha) ----------------
__global__ void edge_alpha_max_kernel(const int* __restrict__ src, const int* __restrict__ dst,
                                      const float* __restrict__ a_src,
                                      const float* __restrict__ a_dst,
                                      float* __restrict__ amax) {
    int e = blockIdx.x * 256 + threadIdx.x;
    if (e >= N_EDGE) return;
    int d = dst[e];
    float al = leaky(a_src[src[e]] + a_dst[d]);
    atomic_max_float(&amax[d], al);
}

// ---------------- edge pass B: segment sum of exp(alpha - amax) ----------------
__global__ void edge_expsum_kernel(const int* __restrict__ src, const int* __restrict__ dst,
                                   const float* __restrict__ a_src,
                                   const float* __restrict__ a_dst,
                                   const float* __restrict__ amax,
                                   float* __restrict__ esum) {
    int e = blockIdx.x * 256 + threadIdx.x;
    if (e >= N_EDGE) return;
    int d = dst[e];
    float al = leaky(a_src[src[e]] + a_dst[d]);
    atomicAdd(&esum[d], __expf(al - amax[d]));
}

// ---------------- edge pass C: weighted scatter h_src[src]*w -> rider_emb[dst] -----
// one wave per edge; float4 (b128) gather, 4 atomic f32 adds per lane into L2-resident
// rider_emb (10000x128 = 5 MB, lives in the 192 MB L2).
__global__ void __launch_bounds__(256) edge_scatter_kernel(
    const int* __restrict__ src, const int* __restrict__ dst,
    const float* __restrict__ a_src, const float* __restrict__ a_dst,
    const float* __restrict__ amax, const float* __restrict__ esum,
    const float* __restrict__ h_src, float* __restrict__ remb) {
    int e = blockIdx.x * 8 + (threadIdx.x >> 5);
    if (e >= N_EDGE) return;
    int lane = threadIdx.x & 31;
    int s = src[e], d = dst[e];
    float al = leaky(a_src[s] + a_dst[d]);
    float w  = __expf(al - amax[d]) / (esum[d] + 1e-16f);
    float4 hv = ((const float4*)(h_src + (size_t)s * HID))[lane];
    float* rp = remb + (size_t)d * HID + lane * 4;
    atomicAdd(rp + 0, hv.x * w);
    atomicAdd(rp + 1, hv.y * w);
    atomicAdd(rp + 2, hv.z * w);
    atomicAdd(rp + 3, hv.w * w);
}

// ---------------- edge pass D: raw score * gate MLP ----------------
__global__ void __launch_bounds__(256) edge_score_kernel(
    const int* __restrict__ src, const int* __restrict__ dst,
    const float* __restrict__ oproj, const float* __restrict__ remb,
    const float* __restrict__ edge_attr,
    const float* __restrict__ Wg1, const float* __restrict__ bg1,
    const float* __restrict__ Wg2, const float* __restrict__ bg2,
    float* __restrict__ out) {
    int e = blockIdx.x * 8 + (threadIdx.x >> 5);
    if (e >= N_EDGE) return;
    int lane = threadIdx.x & 31;
    int s = src[e], d = dst[e];

    // 128-wide dot: 4 elems/lane (b128 loads) + full wave32 xor reduction
    float4 p = ((const float4*)(oproj + (size_t)s * HID))[lane];
    float4 r = ((const float4*)(remb  + (size_t)d * HID))[lane];
    float dot = p.x * r.x + p.y * r.y + p.z * r.z + p.w * r.w;
    dot += __shfl_xor(dot, 16, 32);
    dot += __shfl_xor(dot, 8, 32);
    dot += __shfl_xor(dot, 4, 32);
    dot += __shfl_xor(dot, 2, 32);
    dot += __shfl_xor(dot, 1, 32);

    // gate MLP: hidden unit j computed by lane j (j<16), then reduce + sigmoid
    float g = 0.f;
    if (lane < 16) {
        const float* ea = edge_attr + (size_t)e * 8;
        const float* wr = Wg1 + lane * 8;
        float h = bg1[lane];
#pragma unroll
        for (int k = 0; k < 8; ++k) h = fmaf(ea[k], wr[k], h);
        g = fmaxf(h, 0.f) * Wg2[lane];
    }
    g += __shfl_xor(g, 8, 32);
    g += __shfl_xor(g, 4, 32);
    g += __shfl_xor(g, 2, 32);
    g += __shfl_xor(g, 1, 32);

    if (lane == 0) {
        float z = g + bg2[0];
        out[e] = dot / (1.f + __expf(-z));
    }
}

extern "C" void kernel_launch(void* const* d_in, const int* in_sizes, int n_in,
                              void* d_out, int out_size, void* d_ws, size_t ws_size,
                              hipStream_t stream) {
    (void)in_sizes; (void)n_in; (void)out_size; (void)ws_size;

    const float* x_order   = (const float*)d_in[0];
    const float* x_rider   = (const float*)d_in[1];
    const float* edge_attr = (const float*)d_in[2];
    const int*   esrc      = (const int*)d_in[3];
    const int*   edst      = (const int*)d_in[4];
    const float* W_src     = (const float*)d_in[5];
    const float* W_dst     = (const float*)d_in[6];
    const float* att_src   = (const float*)d_in[7];
    const float* att_dst   = (const float*)d_in[8];
    const float* bias_gat  = (const float*)d_in[9];
    const float* W_proj    = (const float*)d_in[10];
    const float* b_proj    = (const float*)d_in[11];
    const float* W_g1      = (const float*)d_in[12];
    const float* b_g1      = (const float*)d_in[13];
    const float* W_g2      = (const float*)d_in[14];
    const float* b_g2      = (const float*)d_in[15];
    float* out = (float*)d_out;

    float* ws    = (float*)d_ws;
    float* h_src = ws;  ws += (size_t)N_ORDER * HID;   // 12.8M
    float* oproj = ws;  ws += (size_t)N_ORDER * HID;   // 12.8M
    float* a_src = ws;  ws += N_ORDER;
    float* a_dst = ws;  ws += N_RIDER;
    float* amax  = ws;  ws += N_RIDER;
    float* esum  = ws;  ws += N_RIDER;
    float* remb  = ws;  ws += (size_t)N_RIDER * HID;   // 1.28M

    init_kernel<<<(N_RIDER * HID + 255) / 256, 256, 0, stream>>>(
        a_src, a_dst, amax, esum, remb, bias_gat);

    order_gemm_kernel<<<N_ORDER / 16, 256, 0, stream>>>(
        x_order, W_src, W_proj, att_src, b_proj, h_src, oproj, a_src);

    rider_gemm_kernel<<<N_RIDER / 16, 256, 0, stream>>>(
        x_rider, W_dst, att_dst, a_dst);

    edge_alpha_max_kernel<<<(N_EDGE + 255) / 256, 256, 0, stream>>>(
        esrc, edst, a_src, a_dst, amax);

    edge_expsum_kernel<<<(N_EDGE + 255) / 256, 256, 0, stream>>>(
        esrc, edst, a_src, a_dst, amax, esum);

    edge_scatter_kernel<<<(N_EDGE + 7) / 8, 256, 0, stream>>>(
        esrc, edst, a_src, a_dst, amax, esum, h_src, remb);

    edge_score_kernel<<<(N_EDGE + 7) / 8, 256, 0, stream>>>(
        esrc, edst, oproj, remb, edge_attr, W_g1, b_g1, W_g2, b_g2, out);
}